// OurModel_90898687852951
// MI455X (gfx1250) — compile-verified
//
#include <hip/hip_runtime.h>
#include <hip/hip_bf16.h>
#include <math.h>

typedef float v2f __attribute__((ext_vector_type(2)));
typedef float v8f __attribute__((ext_vector_type(8)));

#define Sdim 1024
#define Bdim 128
#define Fdim 202
#define Hdim 100
#define Kdim 19
#define Mrows (Sdim * Bdim)   // 131072 rows of the big GEMMs
#define NT_H 7                // ceil(100/16) n-tiles for H
#define KT_F 51               // ceil(202/4) k-tiles for input proj
#define KT_H 25               // 100/4 k-tiles for recurrence
#define KT_2H 50              // 200/4 k-tiles for output proj

#define WMMA_F32(A, B, C) \
  __builtin_amdgcn_wmma_f32_16x16x4_f32(false, (A), false, (B), (short)0, (C), false, false)

// ---------------------------------------------------------------------------
// Kernel 1: input projection  xp[dir][s*B+b][h] = x @ W_ih^T + b_ih + b_hh
// One wave per 16-wide N tile; W_ih columns live in VGPRs (preloaded once).
// ---------------------------------------------------------------------------
__global__ __launch_bounds__(224) void k_inproj(
    const float* __restrict__ x,
    const float* __restrict__ Wf, const float* __restrict__ bihf, const float* __restrict__ bhhf,
    const float* __restrict__ Wb, const float* __restrict__ bihb, const float* __restrict__ bhhb,
    float* __restrict__ xp, int mtilesPerBlock)
{
  const int wave = threadIdx.x >> 5;      // 0..6 -> n tile
  const int lane = threadIdx.x & 31;
  const int lo = lane & 15, hi = lane >> 4;
  const int dir = blockIdx.y;
  const float* W  = dir ? Wb : Wf;
  const float* bi = dir ? bihb : bihf;
  const float* bh = dir ? bhhb : bhhf;
  const int n = wave * 16 + lo;

  // Preload B (W_ih^T tile columns) into registers: 51 k-tiles x v2f.
  v2f Breg[KT_F];
#pragma unroll
  for (int kt = 0; kt < KT_F; ++kt) {
    int k0 = kt * 4 + hi * 2;
    v2f b = {0.f, 0.f};
    if (n < Hdim && (k0 + 1) < Fdim) b = *(const v2f*)(W + (size_t)n * Fdim + k0);
    Breg[kt] = b;
  }
  const float bias = (n < Hdim) ? (bi[n] + bh[n]) : 0.f;

  const int mt0 = blockIdx.x * mtilesPerBlock;
  for (int mi = 0; mi < mtilesPerBlock; ++mi) {
    const int mt = mt0 + mi;
    const float* xrow = x + (size_t)(mt * 16 + lo) * Fdim;
    v8f acc0, acc1;
#pragma unroll
    for (int r = 0; r < 8; ++r) { acc0[r] = bias; acc1[r] = 0.f; }
#pragma unroll
    for (int kt = 0; kt < KT_F; ++kt) {
      int k0 = kt * 4 + hi * 2;
      v2f a = {0.f, 0.f};
      if ((k0 + 1) < Fdim) a = *(const v2f*)(xrow + k0);
      if (kt & 1) acc1 = WMMA_F32(a, Breg[kt], acc1);
      else        acc0 = WMMA_F32(a, Breg[kt], acc0);
    }
    float* out = xp + ((size_t)dir * Mrows + (size_t)mt * 16) * Hdim;
#pragma unroll
    for (int r = 0; r < 8; ++r) {
      int m = r + 8 * hi;
      if (n < Hdim) out[(size_t)m * Hdim + n] = acc0[r] + acc1[r];
    }
  }
}

// ---------------------------------------------------------------------------
// Kernel 2: recurrence  h_s = tanh(xp_s + h_{s-1} @ W_hh^T)
// grid (8 batch-tiles x 2 dirs); 7 waves (one per N tile). h_{s-1} in LDS,
// W_hh held in VGPRs for all 1024 steps. Two interleaved WMMA accumulators.
// Writes concatenated h: hcat[(s*B+b)][dir*100 + n].
// ---------------------------------------------------------------------------
__global__ __launch_bounds__(224) void k_rnn(
    const float* __restrict__ Whf, const float* __restrict__ Whb,
    const float* __restrict__ xp,      // [2][M][100]
    float* __restrict__ hcat)          // [M][200]
{
  __shared__ float hbuf[16 * 112];     // 16 batch rows x H (padded stride)
  const int wave = threadIdx.x >> 5;   // 0..6 -> n tile
  const int lane = threadIdx.x & 31;
  const int lo = lane & 15, hi = lane >> 4;
  const int dir = blockIdx.y;
  const int mb  = blockIdx.x;          // 0..7 batch tile
  const float* W = dir ? Whb : Whf;
  const int n = wave * 16 + lo;

  v2f Breg[KT_H];
#pragma unroll
  for (int kt = 0; kt < KT_H; ++kt) {
    int k0 = kt * 4 + hi * 2;
    v2f b = {0.f, 0.f};
    if (n < Hdim) b = *(const v2f*)(W + (size_t)n * Hdim + k0);
    Breg[kt] = b;
  }

  for (int i = threadIdx.x; i < 16 * 112; i += 224) hbuf[i] = 0.f;
  __syncthreads();

  const float* xpd = xp + (size_t)dir * Mrows * Hdim;
  for (int ss = 0; ss < Sdim; ++ss) {
    const int s = dir ? (Sdim - 1 - ss) : ss;
    const float* xps = xpd + ((size_t)s * Bdim + mb * 16) * Hdim;
    v8f acc0, acc1;
#pragma unroll
    for (int r = 0; r < 8; ++r) {
      int m = r + 8 * hi;
      acc0[r] = (n < Hdim) ? xps[(size_t)m * Hdim + n] : 0.f;
      acc1[r] = 0.f;
    }
#pragma unroll
    for (int kt = 0; kt < KT_H; ++kt) {
      int k0 = kt * 4 + hi * 2;
      v2f a = *(const v2f*)(&hbuf[lo * 112 + k0]);   // A: rows=batch, k=prev h
      if (kt & 1) acc1 = WMMA_F32(a, Breg[kt], acc1);
      else        acc0 = WMMA_F32(a, Breg[kt], acc0);
    }
    __syncthreads();   // all waves finished reading h_{s-1}
    float* hout = hcat + ((size_t)s * Bdim + mb * 16) * (2 * Hdim) + dir * Hdim;
#pragma unroll
    for (int r = 0; r < 8; ++r) {
      int m = r + 8 * hi;
      float v = tanhf(acc0[r] + acc1[r]);
      if (n < Hdim) {
        hbuf[m * 112 + n] = v;
        hout[(size_t)m * (2 * Hdim) + n] = v;
      }
    }
    __syncthreads();   // h_s visible before next step reads
  }
}

// ---------------------------------------------------------------------------
// Kernel 3: output projection  logits = hcat @ W_out^T + b_out  (K=200, N=19)
// 2 waves per block (N tiles 0..1, padded to 32); W_out in VGPRs.
// ---------------------------------------------------------------------------
__global__ __launch_bounds__(64) void k_outproj(
    const float* __restrict__ hcat,   // [M][200]
    const float* __restrict__ Wout,   // [19][200]
    const float* __restrict__ bout,
    float* __restrict__ logits,       // [M][19]
    int mtilesPerBlock)
{
  const int wave = threadIdx.x >> 5;  // 0..1 -> n tile
  const int lane = threadIdx.x & 31;
  const int lo = lane & 15, hi = lane >> 4;
  const int n = wave * 16 + lo;

  v2f Breg[KT_2H];
#pragma unroll
  for (int kt = 0; kt < KT_2H; ++kt) {
    int k0 = kt * 4 + hi * 2;
    v2f b = {0.f, 0.f};
    if (n < Kdim) b = *(const v2f*)(Wout + (size_t)n * (2 * Hdim) + k0);
    Breg[kt] = b;
  }
  const float bias = (n < Kdim) ? bout[n] : 0.f;

  const int mt0 = blockIdx.x * mtilesPerBlock;
  for (int mi = 0; mi < mtilesPerBlock; ++mi) {
    const int mt = mt0 + mi;
    const float* hrow = hcat + (size_t)(mt * 16 + lo) * (2 * Hdim);
    v8f acc0, acc1;
#pragma unroll
    for (int r = 0; r < 8; ++r) { acc0[r] = bias; acc1[r] = 0.f; }
#pragma unroll
    for (int kt = 0; kt < KT_2H; ++kt) {
      int k0 = kt * 4 + hi * 2;
      v2f a = *(const v2f*)(hrow + k0);
      if (kt & 1) acc1 = WMMA_F32(a, Breg[kt], acc1);
      else        acc0 = WMMA_F32(a, Breg[kt], acc0);
    }
    float* lrow = logits + (size_t)mt * 16 * Kdim;
#pragma unroll
    for (int r = 0; r < 8; ++r) {
      int m = r + 8 * hi;
      if (n < Kdim) lrow[(size_t)m * Kdim + n] = acc0[r] + acc1[r];
    }
  }
}

// ---------------------------------------------------------------------------
// Kernel 4: softmax + Viterbi + backtrace. One wave32 per chain (n = s index),
// lane = tag (19 active). T = B = 128 steps. Cross-lane max via shfl.
// ---------------------------------------------------------------------------
__global__ __launch_bounds__(256) void k_viterbi(
    const float* __restrict__ logits,   // [N=S][T=B][19]
    const float* __restrict__ start_tr,
    const float* __restrict__ end_tr,
    const float* __restrict__ trans,    // [19][19]
    unsigned char* __restrict__ hist,   // [T-1][N][19]
    float* __restrict__ out)            // [N][T]
{
  const int lane = threadIdx.x & 31;
  const int wid  = threadIdx.x >> 5;
  const int n = blockIdx.x * 8 + wid;   // chain id (0..1023)
  const int j = lane;
  const bool act = (j < Kdim);
  const float NEG = -1e30f;

  float tr[Kdim];
#pragma unroll
  for (int i = 0; i < Kdim; ++i) tr[i] = act ? trans[i * Kdim + j] : 0.f;

  // softmax of logits[n, t, :] -> this lane's emission
  auto emrow = [&](int t) -> float {
    float v = act ? logits[((size_t)n * Bdim + t) * Kdim + j] : NEG;
    float m = v;
#pragma unroll
    for (int off = 16; off; off >>= 1) m = fmaxf(m, __shfl_xor(m, off, 32));
    float e = act ? __expf(v - m) : 0.f;
    float ssum = e;
#pragma unroll
    for (int off = 16; off; off >>= 1) ssum += __shfl_xor(ssum, off, 32);
    return e / ssum;
  };

  float score = act ? (start_tr[j] + emrow(0)) : NEG;
  for (int t = 1; t < Bdim; ++t) {
    float em = emrow(t);
    float best = NEG; int bi = 0;
#pragma unroll
    for (int i = 0; i < Kdim; ++i) {
      float c = __shfl(score, i, 32) + tr[i];
      if (c > best) { best = c; bi = i; }   // strict > keeps lowest index on tie
    }
    if (act) hist[((size_t)(t - 1) * Sdim + n) * Kdim + j] = (unsigned char)bi;
    score = act ? (best + em) : NEG;
  }
  score = act ? (score + end_tr[j]) : NEG;

  // argmax over lanes (lowest index on tie)
  float bs = score; int bj = act ? j : Kdim;
#pragma unroll
  for (int off = 16; off; off >>= 1) {
    float os = __shfl_xor(bs, off, 32);
    int   oj = __shfl_xor(bj, off, 32);
    if (os > bs || (os == bs && oj < bj)) { bs = os; bj = oj; }
  }

  if (lane == 0) {
    int tag = bj;
    out[(size_t)n * Bdim + (Bdim - 1)] = (float)tag;
    for (int t = Bdim - 2; t >= 0; --t) {
      tag = hist[((size_t)t * Sdim + n) * Kdim + tag];
      out[(size_t)n * Bdim + t] = (float)tag;
    }
  }
}

// ---------------------------------------------------------------------------
extern "C" void kernel_launch(void* const* d_in, const int* in_sizes, int n_in,
                              void* d_out, int out_size, void* d_ws, size_t ws_size,
                              hipStream_t stream) {
  const float* x      = (const float*)d_in[0];
  const float* W_ih_f = (const float*)d_in[1];
  const float* W_hh_f = (const float*)d_in[2];
  const float* b_ih_f = (const float*)d_in[3];
  const float* b_hh_f = (const float*)d_in[4];
  const float* W_ih_b = (const float*)d_in[5];
  const float* W_hh_b = (const float*)d_in[6];
  const float* b_ih_b = (const float*)d_in[7];
  const float* b_hh_b = (const float*)d_in[8];
  const float* W_out  = (const float*)d_in[9];
  const float* b_out  = (const float*)d_in[10];
  const float* start_tr = (const float*)d_in[11];
  const float* end_tr   = (const float*)d_in[12];
  const float* trans    = (const float*)d_in[13];
  float* out = (float*)d_out;

  // Workspace layout
  float* xp     = (float*)d_ws;                               // 2*M*100
  float* hcat   = xp + (size_t)2 * Mrows * Hdim;              // M*200
  float* logits = hcat + (size_t)Mrows * 2 * Hdim;            // M*19
  unsigned char* hist = (unsigned char*)(logits + (size_t)Mrows * Kdim); // 127*1024*19

  const int MT = Mrows / 16;        // 8192 M-tiles
  // 1) input projection (both directions)
  k_inproj<<<dim3(512, 2), 224, 0, stream>>>(x, W_ih_f, b_ih_f, b_hh_f,
                                             W_ih_b, b_ih_b, b_hh_b,
                                             xp, MT / 512);
  // 2) recurrence (8 batch tiles x 2 directions)
  k_rnn<<<dim3(8, 2), 224, 0, stream>>>(W_hh_f, W_hh_b, xp, hcat);
  // 3) output projection
  k_outproj<<<dim3(512), 64, 0, stream>>>(hcat, W_out, b_out, logits, MT / 512);
  // 4) softmax + Viterbi + backtrace
  k_viterbi<<<dim3(128), 256, 0, stream>>>(logits, start_tr, end_tr, trans, hist, out);
}